// GRCell_22737556865645
// MI455X (gfx1250) — compile-verified
//
#include <hip/hip_runtime.h>

// ---- reference constants (Python-side compile-time constants) ----
#define GAMMA_F 10.0f
#define DT_F    1.0e-4f

// ---- scan decomposition ----
#define CHUNKS_C 2048          // parallel chunks over the T-step scan
#define NPAIR_C  (CHUNKS_C/2)  // pairwise-combined maps (WMMA stage)
#define BURN_C   16384         // Riccati burn-in; IC forgotten at rate >= gamma -> e^-16

typedef __attribute__((ext_vector_type(2)))  float v2f;
typedef __attribute__((ext_vector_type(8)))  float v8f;

// One Euler step of the input-independent Riccati covariance recurrence.
// cov=[[a,b],[b,c]], A=[[-g2,om],[-om,-g2]], S=CtC=s2*[[1,0],[0,0]], D=diag(d00,d11)
__device__ __forceinline__ void cov_step(float& a, float& b, float& c,
                                         float g2, float om, float s2,
                                         float d00, float d11) {
  float da = 2.0f*(om*b - g2*a) + d00 - s2*a*a;
  float db = om*(c - a) - 2.0f*g2*b - s2*a*b;
  float dc = -2.0f*(om*b + g2*c) + d11 - s2*b*b;
  a = fmaf(DT_F, da, a);
  b = fmaf(DT_F, db, b);
  c = fmaf(DT_F, dc, c);
}

// ---------------------------------------------------------------------------
// K1: per-chunk covariance (via burn-in) + composed affine map over the chunk.
// x_{t+1} = E_t x_t + q_t with E_t = I + dt*(A - XiCovC), q_t = s*(a,b)*dy0.
// ---------------------------------------------------------------------------
__global__ void k1_chunk_maps(const float* __restrict__ inputs,
                              const float* __restrict__ omega_p,
                              const float* __restrict__ Cm,
                              const float* __restrict__ Dm,
                              const float* __restrict__ init_state,
                              float* __restrict__ covStart,   // CHUNKS*3
                              float* __restrict__ maps,       // CHUNKS*6
                              int T, int L) {
  int k = blockIdx.x*blockDim.x + threadIdx.x;
  if (k >= CHUNKS_C) return;
  const float om  = omega_p[0];
  const float s   = Cm[0];              // C = s*[[1,0],[0,0]]
  const float s2  = s*s;
  const float d00 = Dm[0], d11 = Dm[3];
  const float g2  = 0.5f*GAMMA_F;

  int t0 = k*L; if (t0 > T) t0 = T;
  int t1 = t0 + L; if (t1 > T) t1 = T;

  // Establish cov at t0: exact from t=0 for early chunks, else burn-in window.
  float a = init_state[2], b = init_state[3], c = init_state[4];
  int tc = (t0 <= BURN_C) ? 0 : (t0 - BURN_C);
  for (; tc < t0; ++tc) cov_step(a, b, c, g2, om, s2, d00, d11);
  covStart[3*k+0] = a; covStart[3*k+1] = b; covStart[3*k+2] = c;

  // Compose affine map (M, v) over [t0, t1): (M,v) <- (E*M, E*v + q)
  float m00=1.f, m01=0.f, m10=0.f, m11=1.f, v0=0.f, v1=0.f;
  for (int t = t0; t < t1; ++t) {
    if (t + 32 < T) __builtin_prefetch(inputs + (t+32)*3 + 1, 0, 1); // global_prefetch_b8
    float dy  = inputs[t*3 + 1];
    float e00 = 1.0f + DT_F*(-g2 - s2*a);
    float e01 = DT_F*om;
    float e10 = DT_F*(-om - s2*b);
    float e11 = 1.0f - DT_F*g2;
    float q0  = s*a*dy;
    float q1  = s*b*dy;
    float n00 = e00*m00 + e01*m10;
    float n01 = e00*m01 + e01*m11;
    float n10 = e10*m00 + e11*m10;
    float n11 = e10*m01 + e11*m11;
    float w0  = e00*v0  + e01*v1  + q0;
    float w1  = e10*v0  + e11*v1  + q1;
    m00=n00; m01=n01; m10=n10; m11=n11; v0=w0; v1=w1;
    cov_step(a, b, c, g2, om, s2, d00, d11);
  }
  float* m = maps + 6*k;
  m[0]=m00; m[1]=m01; m[2]=m10; m[3]=m11; m[4]=v0; m[5]=v1;
}

// ---------------------------------------------------------------------------
// K2: WMMA pairwise composition of chunk maps + serial scan of pair maps.
// Homogeneous 3x3 maps packed block-diagonally (5 per 16x16). Since the A
// operand's row (resp. B's column) is fixed per lane, each lane's nonzero
// elements come from exactly ONE map: load its 6 floats unconditionally and
// build every packed element with pure value-selects (v_cndmask, no EXEC
// divergence -> WMMA sees EXEC all-ones, operand build stays branch-free).
// ---------------------------------------------------------------------------
// Homogeneous element H[r][cc] of map m6=[m00,m01,m10,m11,v0,v1], by value-select.
__device__ __forceinline__ float hsel(const float m6[6], int r, int cc) {
  float row0 = (cc == 0) ? m6[0] : (cc == 1) ? m6[1] : m6[4];
  float row1 = (cc == 0) ? m6[2] : (cc == 1) ? m6[3] : m6[5];
  float row2 = (cc == 2) ? 1.0f : 0.0f;
  return (r == 0) ? row0 : (r == 1) ? row1 : row2;
}

__global__ void __launch_bounds__(256)
k2_combine(const float* __restrict__ maps,
           const float* __restrict__ init_state,
           float* __restrict__ xstart /* CHUNKS*2 */) {
  __shared__ float pm[NPAIR_C * 6];     // 24 KB of pairwise-composed maps
  const int lane   = threadIdx.x & 31;
  const int wave   = threadIdx.x >> 5;
  const int NWAVES = 256 / 32;
  const int NOPS   = (NPAIR_C + 4) / 5; // 5 pair-compositions per wave-matmul

  const int row = lane & 15;            // A row / D column for this lane
  const int col = lane & 15;            // B column (same value, kept for clarity)
  const int rcl = (row < 15) ? row : 14;
  const int blk = rcl / 3;              // the single block this lane touches
  const int rA  = rcl - 3*blk;          // row within block (A)
  const int cB  = rcl - 3*blk;          // col within block (B)

  for (int op = wave; op < NOPS; op += NWAVES) {   // wave-uniform loop bounds
    int basePair = op * 5;
    int pair     = basePair + blk;
    bool pOK     = (pair < NPAIR_C) && (row < 15);

    // Unconditional 6-float map fetches (clamped index; masked by value later).
    int ip = (pair < NPAIR_C) ? pair : (NPAIR_C - 1);
    float mA[6], mB[6];
#pragma unroll
    for (int e = 0; e < 6; ++e) mA[e] = maps[6*(2*ip + 1) + e]; // H_{2p+1}
#pragma unroll
    for (int e = 0; e < 6; ++e) mB[e] = maps[6*(2*ip + 0) + e]; // H_{2p}

    // Chain 4x v_wmma_f32_16x16x4_f32 along K -> full f32 16x16x16 product.
    // A 16x4 group g: lanes0-15 hold K={4g,4g+1}, lanes16-31 K={4g+2,4g+3}.
    v8f acc = {};
#pragma unroll
    for (int g = 0; g < 4; ++g) {
      int kk = 4*g + ((lane >> 4) << 1);
      v2f Av, Bv;
#pragma unroll
      for (int j = 0; j < 2; ++j) {
        int K      = kk + j;
        bool inBlk = (K >= 3*blk) && (K < 3*blk + 3) && (K < 15);
        float av   = (pOK && inBlk) ? hsel(mA, rA, K - 3*blk) : 0.0f;
        float bv   = (pOK && inBlk) ? hsel(mB, K - 3*blk, cB) : 0.0f;
        if (j == 0) { Av.x = av; Bv.x = bv; } else { Av.y = av; Bv.y = bv; }
      }
      acc = __builtin_amdgcn_wmma_f32_16x16x4_f32(false, Av, false, Bv,
                                                  (short)0, acc, false, false);
    }

    // D layout: VGPR r -> rows {r, r+8}, col = lane&15. Scatter diag block to LDS.
#pragma unroll
    for (int r = 0; r < 8; ++r) {
      int drow = r + ((lane >> 4) << 3);
      if (pOK && drow >= 3*blk && drow < 3*blk + 2) {  // rows 0,1 of the block
        int pr = drow - 3*blk, pc = col - 3*blk;       // pc in 0..2
        pm[6*pair + ((pc == 2) ? (4 + pr) : (2*pr + pc))] = acc[r];
      }
    }
  }
  __syncthreads();

  // Serial scan over 1024 pair maps -> x at every chunk start (exact recurrence).
  if (threadIdx.x == 0) {
    float x0 = init_state[0], x1 = init_state[1];
    for (int i = 0; i < NPAIR_C; ++i) {
      int k = 2*i;
      xstart[2*k+0] = x0; xstart[2*k+1] = x1;
      const float* m = maps + 6*k;          // odd chunk start = map_{2i}(x_{2i})
      xstart[2*k+2] = m[0]*x0 + m[1]*x1 + m[4];
      xstart[2*k+3] = m[2]*x0 + m[3]*x1 + m[5];
      const float* p = pm + 6*i;            // even advance via pairwise map
      float y0 = p[0]*x0 + p[1]*x1 + p[4];
      float y1 = p[2]*x0 + p[3]*x1 + p[5];
      x0 = y0; x1 = y1;
    }
  }
}

// ---------------------------------------------------------------------------
// K3: replay each chunk from (covStart, xstart), stream x_new to output.
// ---------------------------------------------------------------------------
__global__ void k3_emit(const float* __restrict__ inputs,
                        const float* __restrict__ omega_p,
                        const float* __restrict__ Cm,
                        const float* __restrict__ Dm,
                        const float* __restrict__ covStart,
                        const float* __restrict__ xstart,
                        float* __restrict__ out,
                        int T, int L) {
  int k = blockIdx.x*blockDim.x + threadIdx.x;
  if (k >= CHUNKS_C) return;
  const float om  = omega_p[0];
  const float s   = Cm[0];
  const float s2  = s*s;
  const float d00 = Dm[0], d11 = Dm[3];
  const float g2  = 0.5f*GAMMA_F;

  int t0 = k*L; if (t0 > T) t0 = T;
  int t1 = t0 + L; if (t1 > T) t1 = T;

  float a  = covStart[3*k+0], b = covStart[3*k+1], c = covStart[3*k+2];
  float x0 = xstart[2*k+0],  x1 = xstart[2*k+1];

  for (int t = t0; t < t1; ++t) {
    if (t + 32 < T) __builtin_prefetch(inputs + (t+32)*3 + 1, 0, 1);
    float dy  = inputs[t*3 + 1];
    float e00 = 1.0f + DT_F*(-g2 - s2*a);
    float e01 = DT_F*om;
    float e10 = DT_F*(-om - s2*b);
    float e11 = 1.0f - DT_F*g2;
    float nx0 = e00*x0 + e01*x1 + s*a*dy;
    float nx1 = e10*x0 + e11*x1 + s*b*dy;
    x0 = nx0; x1 = nx1;
    out[2*t+0] = x0;
    out[2*t+1] = x1;
    cov_step(a, b, c, g2, om, s2, d00, d11);
  }
}

// ---------------------------------------------------------------------------
extern "C" void kernel_launch(void* const* d_in, const int* in_sizes, int n_in,
                              void* d_out, int out_size, void* d_ws, size_t ws_size,
                              hipStream_t stream) {
  const float* inputs  = (const float*)d_in[0];   // (T,3)
  const float* omega_p = (const float*)d_in[1];   // (1,1)
  const float* Cm      = (const float*)d_in[2];   // (2,2)
  const float* Dm      = (const float*)d_in[3];   // (2,2)
  const float* init    = (const float*)d_in[4];   // (5,)
  float* out = (float*)d_out;
  (void)n_in; (void)out_size; (void)ws_size;

  int T = in_sizes[0] / 3;
  int L = (T + CHUNKS_C - 1) / CHUNKS_C;

  float* ws       = (float*)d_ws;
  float* covStart = ws;                       // CHUNKS*3 floats
  float* maps     = covStart + 3*CHUNKS_C;    // CHUNKS*6 floats
  float* xstart   = maps     + 6*CHUNKS_C;    // CHUNKS*2 floats

  dim3 blk(256);
  dim3 grid((CHUNKS_C + 255) / 256);
  k1_chunk_maps<<<grid, blk, 0, stream>>>(inputs, omega_p, Cm, Dm, init,
                                          covStart, maps, T, L);
  k2_combine<<<dim3(1), blk, 0, stream>>>(maps, init, xstart);
  k3_emit<<<grid, blk, 0, stream>>>(inputs, omega_p, Cm, Dm,
                                    covStart, xstart, out, T, L);
}